// LocalSpatialEncoding_82085414961510
// MI455X (gfx1250) — compile-verified
//
#include <hip/hip_runtime.h>
#include <cstdint>

// RandLA-Net LocSE on MI455X (gfx1250, wave32).
// Store-bandwidth bound (1.64 GB out, >= ~70us @ 23.3 TB/s). WMMA f32
// 16x16x4 does the 10->128 pointwise conv; LDS staging turns the WMMA C/D
// layout into coalesced 512B non-temporal float4 stores.

typedef __attribute__((ext_vector_type(2))) float v2f;
typedef __attribute__((ext_vector_type(4))) float v4f;
typedef __attribute__((ext_vector_type(8))) float v8f;

#define KNBR 16
#define DIMF 128
#define RSTR 132   // padded LDS enc row stride (floats): row8 offset % 64 banks = 32 -> no conflict

__global__ __launch_bounds__(256)
void locse_wmma_kernel(const float* __restrict__ coords,    // (N,3)
                       const float* __restrict__ features,  // (N,128)
                       const float* __restrict__ Wc,        // (128,10) row-major
                       const float* __restrict__ bias,      // (128,)
                       const int*   __restrict__ nidx,      // (N,16) int32
                       float* __restrict__ out,             // (N,16,256)
                       int Npts)
{
    __shared__ float sW[DIMF * 10];          // 5 KB, shared by all 8 waves
    __shared__ float sB[DIMF];               // 0.5 KB
    __shared__ float sEnc[8][KNBR * RSTR];   // 8 waves * 16 rows * 132 floats = 66 KB

    const int tid = threadIdx.x;

    // Cooperative stage of W and bias into LDS.
    for (int i = tid; i < DIMF * 10; i += 256) sW[i] = Wc[i];
    if (tid < DIMF) sB[tid] = bias[tid];
    __syncthreads();

    const int  wave = tid >> 5;
    const int  lane = tid & 31;
    const int  lo   = lane & 15;
    const bool hi   = lane >= 16;
    const int  n    = blockIdx.x * 8 + wave;   // one point per wave
    if (n >= Npts) return;                     // wave-uniform exit: EXEC stays all-1s

    // ---- per-row channel vector; row (k) = lo, duplicated across lane halves ----
    const float ox = coords[(size_t)n * 3 + 0];
    const float oy = coords[(size_t)n * 3 + 1];
    const float oz = coords[(size_t)n * 3 + 2];
    const int   jn = nidx[(size_t)n * KNBR + lo];
    const float nx = coords[(size_t)jn * 3 + 0];
    const float ny = coords[(size_t)jn * 3 + 1];
    const float nz = coords[(size_t)jn * 3 + 2];
    const float rx = ox - nx, ry = oy - ny, rz = oz - nz;
    const float dd = sqrtf(rx * rx + ry * ry + rz * rz);
    // channels c0..c9 = {ox,oy,oz, nx,ny,nz, rx,ry,rz, dd}; c10,c11 = 0 (K pad to 12)

    // A fragments (16x4 f32 layout: v.x = K0|K2, v.y = K1|K3 across lane halves)
    v2f a[3];
    a[0].x = hi ? oz : ox;   a[0].y = hi ? nx : oy;     // K {0,1} | {2,3}
    a[1].x = hi ? rx : ny;   a[1].y = hi ? ry : nz;     // K {4,5} | {6,7}
    a[2].x = hi ? 0.f : rz;  a[2].y = hi ? 0.f : dd;    // K {8,9} | {10,11}=0

    float* myEnc = &sEnc[wave][0];
    const int cSel = hi ? 2 : 0;   // K-pair select folded into the LDS address

    // ---- 8 output N-tiles of 16 channels, 3 chained K=4 WMMAs each ----
    #pragma unroll
    for (int t = 0; t < 8; ++t) {
        const int d  = t * 16 + lo;             // output channel for this lane column
        const int dW = d * 10;                  // row base into W (128x10)

        // B fragments (4x16): v.x = K0|K2, v.y = K1|K3. All loads are
        // unconditional (address-select / value-select) so no EXEC toggling.
        v2f b0, b1, b2;
        b0.x = sW[dW + cSel + 0];               // K0|K2
        b0.y = sW[dW + cSel + 1];               // K1|K3
        b1.x = sW[dW + cSel + 4];               // K4|K6
        b1.y = sW[dW + cSel + 5];               // K5|K7
        const float w8 = sW[dW + 8];
        const float w9 = sW[dW + 9];
        b2.x = hi ? 0.f : w8;                   // K8 | K10(=0)
        b2.y = hi ? 0.f : w9;                   // K9 | K11(=0)

        v8f acc = {};
        acc = __builtin_amdgcn_wmma_f32_16x16x4_f32(false, a[0], false, b0,
                                                    (short)0, acc, false, false);
        acc = __builtin_amdgcn_wmma_f32_16x16x4_f32(false, a[1], false, b1,
                                                    (short)0, acc, false, false);
        acc = __builtin_amdgcn_wmma_f32_16x16x4_f32(false, a[2], false, b2,
                                                    (short)0, acc, false, false);

        const float bb = sB[d];
        // C/D layout: VGPR v, lanes0-15 -> row v, lanes16-31 -> row v+8; col = lane%16
        #pragma unroll
        for (int v = 0; v < 8; ++v) {
            const int row = v + (hi ? 8 : 0);
            myEnc[row * RSTR + d] = acc[v] + bb;
        }
    }
    // Same-wave DS ops are in-order: the reads below see the writes above.

    // ---- coalesced non-temporal stores: 64 float4 per output row (enc | feat) ----
    const v4f* f4 = (const v4f*)features;
    const v4f  fv = f4[(size_t)n * 32 + lane];         // this lane's slice of features[n]
    v4f* o4 = (v4f*)out;

    #pragma unroll 4
    for (int k = 0; k < KNBR; ++k) {
        const size_t rowBase = ((size_t)n * KNBR + k) * 64;    // row = 256 floats = 64 x v4f
        const v4f* er4 = (const v4f*)(myEnc + k * RSTR);       // 528B stride: 16B aligned
        const v4f ev = er4[lane];
        __builtin_nontemporal_store(ev, &o4[rowBase + lane]);        // enc[0:128]
        __builtin_nontemporal_store(fv, &o4[rowBase + 32 + lane]);   // feat[128:256]
    }
}

extern "C" void kernel_launch(void* const* d_in, const int* in_sizes, int n_in,
                              void* d_out, int out_size, void* d_ws, size_t ws_size,
                              hipStream_t stream) {
    const float* coords   = (const float*)d_in[0];
    const float* features = (const float*)d_in[1];
    const float* W        = (const float*)d_in[2];
    const float* b        = (const float*)d_in[3];
    const int*   nidx     = (const int*)d_in[4];
    float*       out      = (float*)d_out;

    const int Npts = in_sizes[0] / 3;          // coords is (N,3)
    const int blocks = (Npts + 7) / 8;         // 8 waves/block, 1 point/wave
    locse_wmma_kernel<<<blocks, 256, 0, stream>>>(coords, features, W, b, nidx, out, Npts);
}